// MPO2_12627203850774
// MI455X (gfx1250) — compile-verified
//
#include <hip/hip_runtime.h>

typedef float v2f __attribute__((ext_vector_type(2)));
typedef float v8f __attribute__((ext_vector_type(8)));

// D = A(16x4 f32) * B(4x16 f32) + C, full fp32 accumulate.
#define WMMA_F32(A, B, C) \
  __builtin_amdgcn_wmma_f32_16x16x4_f32(false, (A), false, (B), (short)0, (C), false, false)

// One sample (4096 floats) per wave-pair; 4 samples per 256-thread block.
// LDS per sample: bufA(8192) + bufB(8192) floats, ping-pong between steps.
// Intermediates use column-major tiles [cols][r(16)] so D stores are 2x b128
// and B-fragment K-pairs (p even, p+1) are adjacent -> ds_load_b64.
// Contraction index ordering per step: p = k*2 + a  (k-major, bond a minor).
//   bufA in : [k0(8)][l(512)]                      (x, fixed)
//   bufB l0 : [k1(8)][k2k3(64)][r0=(m0,a0)(16)]    (= [l(512)][r0])
//   bufA l1 : [m0(8)][k2k3(64)][r1=(m1,a1)(16)]
//   bufB l2 : [m0(8)][m1(8)][k3(8)][r2=(m2,a2)(16)]
//   bufA out: [m0m1m2(512)][m3(8)]                 (y, fixed)
__global__ __launch_bounds__(256, 1)
void mpo_tt_kernel(const float* __restrict__ x,
                   const float* __restrict__ W0, const float* __restrict__ W1,
                   const float* __restrict__ W2, const float* __restrict__ W3,
                   float* __restrict__ out, int nSamples)
{
    extern __shared__ float smem[];
    const int tid  = threadIdx.x;
    const int warp = tid >> 5;
    const int lane = tid & 31;
    const int half = warp & 1;      // which half of the 32 tiles this wave does
    const int sloc = warp >> 1;     // sample slot within block (0..3)
    const int col  = lane & 15;     // N index / A-row index
    const int hi   = lane >> 4;     // lane[4]: K slots {0,1} vs {2,3}
    const int koff = hi * 2;

    const int sample = blockIdx.x * 4 + sloc;
    if (sample >= nSamples) return;          // wave-pair-uniform

    float* bufA = smem + sloc * 16384;
    float* bufB = bufA + 8192;

    // ---- Constant A-fragments (A = W', rows r, K slot s = 4f + koff) ----
    const int r = col;
    v2f a0[2], a1[4], a2[4], a3[4];
    {
        const int m = r >> 1, a = r & 1;
#pragma unroll
        for (int f = 0; f < 2; ++f) {            // step0: p = k0 (K=8)
            int k = 4 * f + koff;
            a0[f].x = W0[a * 64 + k * 8 + m];
            a0[f].y = W0[a * 64 + (k + 1) * 8 + m];
        }
#pragma unroll
        for (int f = 0; f < 4; ++f) {            // steps1..3: p = k*2 + a_in (K=16)
            int s = 4 * f + koff;                // even -> a_in = 0; s+1 -> a_in = 1
            int k = s >> 1;
            int idx = a * 64 + k * 8 + m;        // W[a_in][a_out][k][m], a_in stride 128
            a1[f].x = W1[idx];        a1[f].y = W1[idx + 128];
            a2[f].x = W2[idx];        a2[f].y = W2[idx + 128];
            int idx3 = k * 8 + r;                // W3[a_in][0][k][m=r], a_in stride 64
            a3[f].x = (r < 8) ? W3[idx3]      : 0.0f;   // A rows 8..15 zero
            a3[f].y = (r < 8) ? W3[idx3 + 64] : 0.0f;
        }
    }

    // ---- Stage x -> LDS bufA (float4, 64 threads per sample) ----
    const int ptid = half * 32 + lane;           // 0..63 within the wave pair
    {
        const float4* xs  = reinterpret_cast<const float4*>(x + (size_t)sample * 4096);
        float4*       dst = reinterpret_cast<float4*>(bufA);
#pragma unroll
        for (int i = 0; i < 16; ++i) dst[ptid + 64 * i] = xs[ptid + 64 * i];
    }
    __syncthreads();

    // B 4x16 f32: VGPR0 lanes0-15 K=0 / lanes16-31 K=2; VGPR1 K=1 / K=3.
    // D 16x16 f32: VGPR g, lanes0-15 M=g / lanes16-31 M=g+8, N=lane&15.

    // ---- Step 0: contract k0. in bufA[k0*512 + l] -> bufB[l*16 + r0] ----
    for (int t = half * 16; t < half * 16 + 16; ++t) {
        const int c = t * 16 + col;              // column = l (0..511)
        v8f acc = {};
#pragma unroll
        for (int f = 0; f < 2; ++f) {
            int k = 4 * f + koff;
            v2f b;                               // k0 stride 512: two b32 loads
            b.x = bufA[k * 512 + c];
            b.y = bufA[k * 512 + 512 + c];
            acc = WMMA_F32(a0[f], b, acc);
        }
        float4* st = reinterpret_cast<float4*>(&bufB[c * 16 + 8 * hi]);
        st[0] = make_float4(acc[0], acc[1], acc[2], acc[3]);
        st[1] = make_float4(acc[4], acc[5], acc[6], acc[7]);
    }
    __syncthreads();

    // ---- Step 1: contract (k1,a0). in bufB[(k1*64+l2)*16 + m0*2 + a0] ----
    for (int t = half * 16; t < half * 16 + 16; ++t) {
        const int cg = t * 16 + col;             // column = (m0, l2)
        const int m0 = cg >> 6, l2 = cg & 63;
        v8f acc = {};
#pragma unroll
        for (int f = 0; f < 4; ++f) {
            int k1 = (4 * f + koff) >> 1;        // pair (a0=0, a0=1) adjacent -> b64
            v2f b = *reinterpret_cast<const v2f*>(&bufB[(k1 * 64 + l2) * 16 + m0 * 2]);
            acc = WMMA_F32(a1[f], b, acc);
        }
        float4* st = reinterpret_cast<float4*>(&bufA[(m0 * 64 + l2) * 16 + 8 * hi]);
        st[0] = make_float4(acc[0], acc[1], acc[2], acc[3]);
        st[1] = make_float4(acc[4], acc[5], acc[6], acc[7]);
    }
    __syncthreads();

    // ---- Step 2: contract (k2,a1). in bufA[(m0*64+k2*8+k3)*16 + m1*2 + a1] ----
    for (int t = half * 16; t < half * 16 + 16; ++t) {
        const int cg = t * 16 + col;             // column = (m0, m1, k3)
        const int m0 = cg >> 6, m1 = (cg >> 3) & 7, k3 = cg & 7;
        v8f acc = {};
#pragma unroll
        for (int f = 0; f < 4; ++f) {
            int k2 = (4 * f + koff) >> 1;
            v2f b = *reinterpret_cast<const v2f*>(&bufA[(m0 * 64 + k2 * 8 + k3) * 16 + m1 * 2]);
            acc = WMMA_F32(a2[f], b, acc);
        }
        float4* st = reinterpret_cast<float4*>(&bufB[cg * 16 + 8 * hi]);
        st[0] = make_float4(acc[0], acc[1], acc[2], acc[3]);
        st[1] = make_float4(acc[4], acc[5], acc[6], acc[7]);
    }
    __syncthreads();

    // ---- Step 3: contract (k3,a2). in bufB[((m0*8+m1)*8+k3)*16 + m2*2 + a2] ----
    for (int t = half * 16; t < half * 16 + 16; ++t) {
        const int cg = t * 16 + col;             // column = i3 = (m0, m1, m2)
        const int m01 = cg >> 3, m2 = cg & 7;
        v8f acc = {};
#pragma unroll
        for (int f = 0; f < 4; ++f) {
            int k3 = (4 * f + koff) >> 1;
            v2f b = *reinterpret_cast<const v2f*>(&bufB[(m01 * 8 + k3) * 16 + m2 * 2]);
            acc = WMMA_F32(a3[f], b, acc);
        }
        if (hi == 0) {                           // rows 8..15 are zero padding
            float4* st = reinterpret_cast<float4*>(&bufA[cg * 8]);
            st[0] = make_float4(acc[0], acc[1], acc[2], acc[3]);
            st[1] = make_float4(acc[4], acc[5], acc[6], acc[7]);
        }
    }
    __syncthreads();

    // ---- Store y (float4) ----
    {
        float4*       od  = reinterpret_cast<float4*>(out + (size_t)sample * 4096);
        const float4* src = reinterpret_cast<const float4*>(bufA);
#pragma unroll
        for (int i = 0; i < 16; ++i) od[ptid + 64 * i] = src[ptid + 64 * i];
    }
}

extern "C" void kernel_launch(void* const* d_in, const int* in_sizes, int n_in,
                              void* d_out, int out_size, void* d_ws, size_t ws_size,
                              hipStream_t stream)
{
    const float* x  = (const float*)d_in[0];
    const float* W0 = (const float*)d_in[1];
    const float* W1 = (const float*)d_in[2];
    const float* W2 = (const float*)d_in[3];
    const float* W3 = (const float*)d_in[4];
    float* out = (float*)d_out;

    const int nSamples = in_sizes[0] / 4096;          // B*S = 8192
    const int blocks   = (nSamples + 3) / 4;          // 4 samples per block
    const size_t shmem = 4 * 16384 * sizeof(float);   // 256 KB dynamic LDS

    (void)hipFuncSetAttribute(reinterpret_cast<const void*>(mpo_tt_kernel),
                              hipFuncAttributeMaxDynamicSharedMemorySize, (int)shmem);
    mpo_tt_kernel<<<blocks, 256, shmem, stream>>>(x, W0, W1, W2, W3, out, nSamples);
}